// CustomCTC_33114197852164
// MI455X (gfx1250) — compile-verified
//
#include <hip/hip_runtime.h>
#include <math.h>

// Problem constants (reference: T=1024, B=32, C=1024, S=128)
#define C_DIM 1024
#define S_DIM 128
#define SE_DIM (2 * S_DIM + 1)   // 257 extended label positions
#define SPL 9                    // slots per lane: ceil(257/32)
#define NSLOT (32 * SPL)         // 288
#define NBUF 8                   // LDS emission buffers (ring)
#define PFD 4                    // async prefetch distance (iterations)
#define PFL2 16                  // L2 warm-up prefetch distance
#define NEGV (-1e30f)
#define LOG2E 1.4426950408889634f
#define LN2 0.6931471805599453f

#define WAITA(n) asm volatile("s_wait_asynccnt " #n ::: "memory")

// Raw 32-bit LDS byte offset of a __shared__ object (async-DMA vdst operand).
__device__ __forceinline__ unsigned lds_off(void* p) {
  return (unsigned)(unsigned long long)(__attribute__((address_space(3))) char*)p;
}

// Issue 9 scattered async global->LDS gathers (one per alpha slot of this lane).
__device__ __forceinline__ void issue_async_gather(const float* row,
                                                   unsigned ldsbase,
                                                   const int* coff) {
#pragma unroll
  for (int j = 0; j < SPL; ++j)
    asm volatile("global_load_async_to_lds_b32 %0, %1, %2"
                 :: "v"(ldsbase + 4u * (unsigned)j), "v"(coff[j]), "s"(row)
                 : "memory");
}

// One wave (32 lanes) per batch sample. Alpha lives in registers (9 per lane,
// blocked layout s = lane*9 + j), kept in the LOG2 domain so the 3-way
// log-sum-exp uses raw v_exp_f32 / v_log_f32 (native base-2 transcendentals).
// Cross-lane neighbors via wave32 shuffles. Emission gathers are prefetched
// PFD steps ahead via CDNA5 async global->LDS DMA (ASYNCcnt), with an L2
// warm-up prefetch PFL2 steps ahead. No barriers in the hot loop.
__global__ __launch_bounds__(32) void ctc_alpha_kernel(
    const float* __restrict__ lp,      // (T, B, C) log-softmax (natural log)
    const int* __restrict__ labels,    // (B, S)
    const int* __restrict__ in_len,    // (B,)
    const int* __restrict__ tg_len,    // (B,)
    float* __restrict__ ws,            // (B,) per-sample NLL out
    int T, int B)
{
  __shared__ float ebuf[NBUF][NSLOT]; // ring of gathered-emission buffers
  const int b = blockIdx.x;
  const int lane = threadIdx.x;
  const int Lb = tg_len[b];
  const int Tb = in_len[b];

  // Per-slot static metadata: class byte-offset, valid mask, allow-skip mask.
  int coff[SPL];
  unsigned validM = 0u, allowM = 0u;
#pragma unroll
  for (int j = 0; j < SPL; ++j) {
    const int s = lane * SPL + j;
    int cls = 0;                                   // even s -> blank(0)
    if (s < SE_DIM && (s & 1)) cls = labels[b * S_DIM + ((s - 1) >> 1)];
    coff[j] = cls << 2;                            // byte offset into class row
    if (s < 2 * Lb + 1) validM |= 1u << j;
    if ((s & 1) && s >= 3 && s < SE_DIM) {         // non-blank, s>=2
      const int pc = labels[b * S_DIM + ((s - 3) >> 1)];
      if (cls != pc) allowM |= 1u << j;            // ext[s] != ext[s-2]
    }
  }

  // t = 0 initialization (log2 domain): only s=0 and s=1 are live.
  float a[SPL];
#pragma unroll
  for (int j = 0; j < SPL; ++j) a[j] = NEGV;
  if (lane == 0) {
    const float* row0 = lp + (size_t)b * C_DIM;    // t=0 row for sample b
    a[0] = row0[0] * LOG2E;
    if (Lb >= 1) a[1] = row0[labels[b * S_DIM]] * LOG2E;
  }

  const unsigned lbase = lds_off(&ebuf[0][lane * SPL]);   // + buf*NSLOT*4

  // Prime the pipeline: async-prefetch emissions for t = 1 .. min(PFD, T-1).
  for (int tp = 1; tp <= PFD && tp < T; ++tp) {
    const float* row = lp + ((size_t)tp * B + b) * C_DIM;
    issue_async_gather(row, lbase + (unsigned)(tp & (NBUF - 1)) * (NSLOT * 4u),
                       coff);
  }

  for (int t = 1; t < T; ++t) {
    // Warm L2 for the row PFL2 steps ahead: 32 lanes x 128B = the full 4KB row.
    if (t + PFL2 < T) {
      const float* prow = lp + ((size_t)(t + PFL2) * B + b) * C_DIM + lane * 32;
      __builtin_prefetch(prow, 0, 1);
    }

    // Keep PFD batches of 9 async gathers in flight; wait until step-t's 9
    // have landed (async loads complete in order).
    if (t + PFD < T) {
      const float* row = lp + ((size_t)(t + PFD) * B + b) * C_DIM;
      issue_async_gather(
          row, lbase + (unsigned)((t + PFD) & (NBUF - 1)) * (NSLOT * 4u), coff);
      WAITA(36);                                   // 9 * PFD
    } else if (t + 3 < T) { WAITA(27); }
    else if (t + 2 < T)   { WAITA(18); }
    else if (t + 1 < T)   { WAITA(9);  }
    else                  { WAITA(0);  }

    // Emissions for step t from LDS, scaled into the log2 domain (one mul).
    const float* eb = &ebuf[t & (NBUF - 1)][lane * SPL];
    float emit[SPL];
#pragma unroll
    for (int j = 0; j < SPL; ++j) emit[j] = eb[j] * LOG2E;

    // Two boundary alphas from the previous lane (old slots 8 and 7).
    float p8 = __shfl_up(a[SPL - 1], 1, 32);
    float p7 = __shfl_up(a[SPL - 2], 1, 32);
    if (lane == 0) { p8 = NEGV; p7 = NEGV; }

    const bool act = (t < Tb);                     // freeze past input length
    float na[SPL];
#pragma unroll
    for (int j = 0; j < SPL; ++j) {
      const float a0 = a[j];
      const float a1 = (j == 0) ? p8 : a[j - 1];
      const float a2r = (j == 0) ? p7 : ((j == 1) ? p8 : a[j - 2]);
      const float a2 = ((allowM >> j) & 1u) ? a2r : NEGV;
      const float m = fmaxf(a0, fmaxf(a1, a2));
      // log2-sum-exp2: native v_exp_f32 / v_log_f32, no range-reduction code.
      const float sum = __builtin_amdgcn_exp2f(a0 - m) +
                        __builtin_amdgcn_exp2f(a1 - m) +
                        __builtin_amdgcn_exp2f(a2 - m);
      const float lse = m + __builtin_amdgcn_logf(sum);
      const float nv = ((validM >> j) & 1u) ? (lse + emit[j]) : NEGV;
      na[j] = act ? nv : a0;
    }
#pragma unroll
    for (int j = 0; j < SPL; ++j) a[j] = na[j];
  }

  // Stage final alpha to LDS; lane 0 extracts the two terminal states.
  __syncthreads();
#pragma unroll
  for (int j = 0; j < SPL; ++j) ebuf[0][lane * SPL + j] = a[j];
  __syncthreads();

  if (lane == 0) {
    const int s1 = 2 * Lb;
    const int s2 = (2 * Lb - 1 >= 0) ? (2 * Lb - 1) : (SE_DIM - 1);
    const float l1 = ebuf[0][s1];                  // log2 domain
    const float l2 = ebuf[0][s2];
    const float m = fmaxf(l1, l2);
    const float lse2 = m + __builtin_amdgcn_logf(
        __builtin_amdgcn_exp2f(l1 - m) + __builtin_amdgcn_exp2f(l2 - m));
    float per = -(lse2 * LN2);                     // back to natural log
    if (per > 1e29f) per = 0.0f;                   // zero_infinity
    ws[b] = per;
  }
}

// Sum per-sample NLLs and apply focal transform:
// out = ALPHA * (1 - exp(-ctc))^GAMMA * ctc, ALPHA=0.25, GAMMA=0.5.
__global__ __launch_bounds__(32) void ctc_focal_reduce(
    const float* __restrict__ ws, float* __restrict__ out, int B)
{
  const int l = threadIdx.x;
  float v = 0.0f;
  for (int i = l; i < B; i += 32) v += ws[i];
#pragma unroll
  for (int off = 16; off > 0; off >>= 1) v += __shfl_down(v, off, 32);
  if (l == 0) {
    const float ctc = v;
    const float p = expf(-ctc);
    const float om = fmaxf(1.0f - p, 0.0f);
    out[0] = 0.25f * sqrtf(om) * ctc;
  }
}

extern "C" void kernel_launch(void* const* d_in, const int* in_sizes, int n_in,
                              void* d_out, int out_size, void* d_ws, size_t ws_size,
                              hipStream_t stream) {
  const float* lp     = (const float*)d_in[0];   // (T,B,C) float32
  const int*   labels = (const int*)d_in[1];     // (B,S) int32
  const int*   il     = (const int*)d_in[2];     // (B,) int32
  const int*   tl     = (const int*)d_in[3];     // (B,) int32
  float* out = (float*)d_out;
  float* ws  = (float*)d_ws;                     // B floats of scratch

  const int B = in_sizes[2];
  const int T = in_sizes[0] / (B * C_DIM);

  ctc_alpha_kernel<<<B, 32, 0, stream>>>(lp, labels, il, tl, ws, T, B);
  ctc_focal_reduce<<<1, 32, 0, stream>>>(ws, out, B);
}